// GNNPPOPolicy_3770981286028
// MI455X (gfx1250) — compile-verified
//
#include <hip/hip_runtime.h>
#include <math.h>

typedef __attribute__((ext_vector_type(2))) float v2f;
typedef __attribute__((ext_vector_type(8))) float v8f;

#define D_INF 128
#define D_HID 128
#define D_OUTP 10
#define GEMM_WAVES 4
#define LDS_MAXK 256
#define TILE_M 32

#if __has_builtin(__builtin_amdgcn_global_load_async_to_lds_b128)
#define USE_ASYNC_LDS 1
#endif

// exact parameter types expected by the async-to-LDS builtin (from hipcc diagnostic):
//   arg0: int __attribute__((vector_size(16))) addrspace(1)*   (global src)
//   arg1: same vector type addrspace(3)*                        (LDS dst)
typedef int v4i __attribute__((vector_size(16)));
typedef __attribute__((address_space(1))) v4i* gvec_t;
typedef __attribute__((address_space(3))) v4i* lvec_t;

// -------------------- degree / symmetric normalization --------------------
__global__ void deg_init(float* deg, int N) {
  int i = blockIdx.x * blockDim.x + threadIdx.x;
  if (i < N) deg[i] = 1.0f;  // self-loop contributes 1 to every node's degree
}
__global__ void deg_accum(const int* __restrict__ ei, float* deg, int E) {
  int e = blockIdx.x * blockDim.x + threadIdx.x;
  if (e < E) atomicAdd(&deg[ei[E + e]], 1.0f);  // col = destination
}
__global__ void deg_rsqrt(float* deg, int N) {
  int i = blockIdx.x * blockDim.x + threadIdx.x;
  if (i < N) deg[i] = rsqrtf(deg[i]);  // deg >= 1 always (self loops)
}

// -------------------- WMMA fp32 GEMM: out = [relu?](A) @ W + bias --------------------
// A: [N x Din] row-major, W: [Din x Dout] row-major, out: [N x Dout].
// Block = 128 threads = 4 waves; block owns a 32-row (2 M-tile) slab of A staged in
// LDS (async DMA when available); each wave sweeps 16-column tiles, sharing each B
// fragment across both M-tiles (2 WMMAs per B load pair). RELU is compile-time so
// the K loop is branch-free; all hot addresses are running pointers/offsets.
template <bool RELU>
__global__ __launch_bounds__(128) void gemm_wmma(
    const float* __restrict__ A, const float* __restrict__ W,
    const float* __restrict__ bias, float* __restrict__ out,
    int Din, int Dout)
{
  __shared__ float lds[TILE_M * (LDS_MAXK + 4)];
  const int LDSTR = Din + 4;                 // padded stride: 4-bank skew per row
  const int row0  = blockIdx.x * TILE_M;

  // stage the 32 x Din tile into LDS (raw copy; ReLU applied at fragment load)
  {
    const int c4n = Din >> 2;
    const int nf4 = TILE_M * c4n;
#ifdef USE_ASYNC_LDS
    for (int f = threadIdx.x; f < nf4; f += blockDim.x) {
      int r = f / c4n, c4 = f - r * c4n;
      __builtin_amdgcn_global_load_async_to_lds_b128(
          (gvec_t)(A + (size_t)(row0 + r) * Din + (c4 << 2)),
          (lvec_t)&lds[r * LDSTR + (c4 << 2)], 0, 0);
    }
#if __has_builtin(__builtin_amdgcn_s_wait_asynccnt)
    __builtin_amdgcn_s_wait_asynccnt(0);
#else
    asm volatile("s_wait_asynccnt 0" ::: "memory");
#endif
#else
    for (int f = threadIdx.x; f < nf4; f += blockDim.x) {
      int r = f / c4n, c4 = f - r * c4n;
      float4 v = ((const float4*)(A + (size_t)(row0 + r) * Din))[c4];
      *(float4*)&lds[r * LDSTR + (c4 << 2)] = v;
    }
#endif
  }
  __syncthreads();

  const int wave = threadIdx.x >> 5;
  const int lane = threadIdx.x & 31;
  const int m    = lane & 15;                // A row / B,C,D column within tile
  const int kb   = (lane >> 4) << 1;         // K sub-offset: 0 (lanes 0-15) or 2 (16-31)
  const int ntiles = Dout >> 4;

  for (int ct = wave; ct < ntiles; ct += GEMM_WAVES) {
    const int ncol = (ct << 4) + m;
    v8f acc0 = {0.f, 0.f, 0.f, 0.f, 0.f, 0.f, 0.f, 0.f};
    v8f acc1 = {0.f, 0.f, 0.f, 0.f, 0.f, 0.f, 0.f, 0.f};
#if __has_builtin(__builtin_amdgcn_wmma_f32_16x16x4_f32)
    const float* wp = W + (size_t)kb * Dout + ncol;   // running B pointer
    int la0 = m        * LDSTR + kb;                  // running LDS offsets
    int la1 = (m + 16) * LDSTR + kb;
    for (int k0 = 0; k0 < Din; k0 += 4) {
      v2f b;
      b.x = wp[0];                                    // B[k0+kb  ][ncol]
      b.y = wp[Dout];                                 // B[k0+kb+1][ncol]
      v2f a0 = *(const v2f*)&lds[la0];
      v2f a1 = *(const v2f*)&lds[la1];
      if (RELU) {
        a0.x = fmaxf(a0.x, 0.f); a0.y = fmaxf(a0.y, 0.f);
        a1.x = fmaxf(a1.x, 0.f); a1.y = fmaxf(a1.y, 0.f);
      }
      acc0 = __builtin_amdgcn_wmma_f32_16x16x4_f32(
          false, a0, false, b, (short)0, acc0, false, false);
      acc1 = __builtin_amdgcn_wmma_f32_16x16x4_f32(
          false, a1, false, b, (short)0, acc1, false, false);
      wp  += (size_t)4 * Dout;
      la0 += 4; la1 += 4;
    }
#else
    // FMA fallback with identical output fragment layout
    for (int k = 0; k < Din; ++k) {
      float wv = W[(size_t)k * Dout + ncol];
      #pragma unroll
      for (int j = 0; j < 8; ++j) {
        int r = ((lane >> 4) << 3) + j;
        float x0 = lds[r        * LDSTR + k];
        float x1 = lds[(r + 16) * LDSTR + k];
        if (RELU) { x0 = fmaxf(x0, 0.f); x1 = fmaxf(x1, 0.f); }
        acc0[j] += x0 * wv;
        acc1[j] += x1 * wv;
      }
    }
#endif
    const float bv = bias ? bias[ncol] : 0.f;
    // C/D fragment: M = j + 8*lane[4]; walk output with a running pointer
    float* op = out + (size_t)(row0 + ((lane >> 4) << 3)) * Dout + ncol;
    const size_t half = (size_t)16 * Dout;
    #pragma unroll
    for (int j = 0; j < 8; ++j) {
      op[0]    = acc0[j] + bv;
      op[half] = acc1[j] + bv;
      op += Dout;
    }
  }
}

// -------------------- GCN aggregation --------------------
// out = bias + hW * dinv^2  (self-loop message fused with bias init)
__global__ void agg_init(const float* __restrict__ hW, const float* __restrict__ dinv,
                         const float* __restrict__ bias, float* __restrict__ out, int N)
{
  int i = blockIdx.x * blockDim.x + threadIdx.x;
  if (i < N * D_HID) {
    int node = i >> 7;       // / 128
    int d    = i & 127;
    float di = dinv[node];
    out[i] = bias[d] + hW[i] * di * di;
  }
}

// per-edge scatter-add: out[col] += hW[row] * dinv[row]*dinv[col]
__global__ __launch_bounds__(128) void agg_edges(
    const float* __restrict__ hW, const int* __restrict__ ei,
    const float* __restrict__ dinv, float* __restrict__ out, int E)
{
  const int d = threadIdx.x;            // 128 features, one per thread
  for (int e = blockIdx.x; e < E; e += gridDim.x) {
    int r = ei[e], c = ei[E + e];
    float nrm = dinv[r] * dinv[c];
    atomicAdd(&out[(size_t)c * D_HID + d], hW[(size_t)r * D_HID + d] * nrm);
  }
}

// -------------------- final heads: softmax(logits) and value --------------------
// one wave32 per node
__global__ __launch_bounds__(128) void heads_kernel(
    const float* __restrict__ actH, const float* __restrict__ criH,
    const float* __restrict__ A3, const float* __restrict__ a3,
    const float* __restrict__ V3, const float* __restrict__ v3,
    float* __restrict__ probs, float* __restrict__ vals, int N)
{
  int node = blockIdx.x * (blockDim.x >> 5) + (threadIdx.x >> 5);
  if (node >= N) return;
  const int lane = threadIdx.x & 31;
  const int k = lane << 2;  // each lane covers 4 of 128 features

  // ---- actor logits + softmax ----
  float4 ar = ((const float4*)(actH + (size_t)node * D_HID))[lane];
  ar.x = fmaxf(ar.x, 0.f); ar.y = fmaxf(ar.y, 0.f);
  ar.z = fmaxf(ar.z, 0.f); ar.w = fmaxf(ar.w, 0.f);

  float logit[D_OUTP];
  #pragma unroll
  for (int j = 0; j < D_OUTP; ++j) {
    float p = ar.x * A3[(k + 0) * D_OUTP + j] + ar.y * A3[(k + 1) * D_OUTP + j]
            + ar.z * A3[(k + 2) * D_OUTP + j] + ar.w * A3[(k + 3) * D_OUTP + j];
    #pragma unroll
    for (int off = 16; off >= 1; off >>= 1) p += __shfl_xor(p, off, 32);
    logit[j] = p + a3[j];
  }
  float mx = logit[0];
  #pragma unroll
  for (int j = 1; j < D_OUTP; ++j) mx = fmaxf(mx, logit[j]);
  float ex[D_OUTP], s = 0.f;
  #pragma unroll
  for (int j = 0; j < D_OUTP; ++j) { ex[j] = __expf(logit[j] - mx); s += ex[j]; }
  float inv = 1.f / s;
  if (lane == 0) {
    #pragma unroll
    for (int j = 0; j < D_OUTP; ++j) probs[(size_t)node * D_OUTP + j] = ex[j] * inv;
  }

  // ---- critic value ----
  float4 vr = ((const float4*)(criH + (size_t)node * D_HID))[lane];
  vr.x = fmaxf(vr.x, 0.f); vr.y = fmaxf(vr.y, 0.f);
  vr.z = fmaxf(vr.z, 0.f); vr.w = fmaxf(vr.w, 0.f);
  float q = vr.x * V3[k] + vr.y * V3[k + 1] + vr.z * V3[k + 2] + vr.w * V3[k + 3];
  #pragma unroll
  for (int off = 16; off >= 1; off >>= 1) q += __shfl_xor(q, off, 32);
  if (lane == 0) vals[node] = q + v3[0];
}

// -------------------- host orchestration --------------------
extern "C" void kernel_launch(void* const* d_in, const int* in_sizes, int n_in,
                              void* d_out, int out_size, void* d_ws, size_t ws_size,
                              hipStream_t stream) {
  (void)n_in; (void)out_size; (void)ws_size;
  const float* x  = (const float*)d_in[0];
  const int*   ei = (const int*)  d_in[1];
  const float* W1 = (const float*)d_in[2];  const float* b1 = (const float*)d_in[3];
  const float* W2 = (const float*)d_in[4];  const float* b2 = (const float*)d_in[5];
  const float* W3 = (const float*)d_in[6];  const float* b3 = (const float*)d_in[7];
  const float* A1 = (const float*)d_in[8];  const float* a1 = (const float*)d_in[9];
  const float* A2 = (const float*)d_in[10]; const float* a2 = (const float*)d_in[11];
  const float* A3 = (const float*)d_in[12]; const float* a3 = (const float*)d_in[13];
  const float* V1 = (const float*)d_in[14]; const float* v1 = (const float*)d_in[15];
  const float* V2 = (const float*)d_in[16]; const float* v2 = (const float*)d_in[17];
  const float* V3 = (const float*)d_in[18]; const float* v3 = (const float*)d_in[19];

  const int N = in_sizes[0] / D_INF;   // 100000 (divisible by 32)
  const int E = in_sizes[1] / 2;       // 1600000

  // workspace layout (floats), offsets padded to 256-float boundaries
  float* ws = (float*)d_ws;
  size_t off = 0;
  float* dinv = ws + off; off += (((size_t)N + 255) & ~(size_t)255);
  float* buf0 = ws + off; off += (size_t)N * 256;   // transforms / head hidden-1
  float* buf1 = ws + off; off += (size_t)N * 128;   // aggregated h / critic hidden-2
  float* buf2 = ws + off; off += (size_t)N * 128;   // actor hidden-2

  const dim3 B256(256);
  deg_init <<<(N + 255) / 256, B256, 0, stream>>>(dinv, N);
  deg_accum<<<(E + 255) / 256, B256, 0, stream>>>(ei, dinv, E);
  deg_rsqrt<<<(N + 255) / 256, B256, 0, stream>>>(dinv, N);

  const int rowTiles = N / TILE_M;
  const int aggGrid  = (int)(((size_t)N * D_HID + 255) / 256);

  // ---- GCN layer 1: h = relu(agg(x @ W1) + b1) (relu deferred to consumer) ----
  gemm_wmma<false><<<rowTiles, 128, 0, stream>>>(x,    W1, nullptr, buf0, D_INF, D_HID);
  agg_init <<<aggGrid, B256, 0, stream>>>(buf0, dinv, b1, buf1, N);
  agg_edges<<<E, 128, 0, stream>>>(buf0, ei, dinv, buf1, E);
  // ---- GCN layer 2 ----
  gemm_wmma<true><<<rowTiles, 128, 0, stream>>>(buf1, W2, nullptr, buf0, D_HID, D_HID);
  agg_init <<<aggGrid, B256, 0, stream>>>(buf0, dinv, b2, buf1, N);
  agg_edges<<<E, 128, 0, stream>>>(buf0, ei, dinv, buf1, E);
  // ---- GCN layer 3 ----
  gemm_wmma<true><<<rowTiles, 128, 0, stream>>>(buf1, W3, nullptr, buf0, D_HID, D_HID);
  agg_init <<<aggGrid, B256, 0, stream>>>(buf0, dinv, b3, buf1, N);
  agg_edges<<<E, 128, 0, stream>>>(buf0, ei, dinv, buf1, E);
  // buf1 now holds h3 (pre-ReLU); heads apply ReLU on load.

  // ---- actor MLP ----
  gemm_wmma<true><<<rowTiles, 128, 0, stream>>>(buf1, A1, a1, buf0, D_HID, 2 * D_HID);
  gemm_wmma<true><<<rowTiles, 128, 0, stream>>>(buf0, A2, a2, buf2, 2 * D_HID, D_HID);
  // ---- critic MLP (buf0 free again; buf1 consumed after first gemm) ----
  gemm_wmma<true><<<rowTiles, 128, 0, stream>>>(buf1, V1, v1, buf0, D_HID, 2 * D_HID);
  gemm_wmma<true><<<rowTiles, 128, 0, stream>>>(buf0, V2, v2, buf1, 2 * D_HID, D_HID);

  // ---- final heads: softmax probs [N x 10] then values [N] ----
  float* probs = (float*)d_out;
  float* vals  = probs + (size_t)N * D_OUTP;
  heads_kernel<<<(N + 3) / 4, 128, 0, stream>>>(buf2, buf1, A3, a3, V3, v3, probs, vals, N);
}